// TinyTemporalEventModel_64707977282164
// MI455X (gfx1250) — compile-verified
//
#include <hip/hip_runtime.h>
#include <hip/hip_bf16.h>

// Problem constants (from reference)
#define B_   128
#define N_   2048
#define E_   32768
#define C_   4
#define TC_  4
#define H_   2
#define L_   2
#define LOG2E_ 15   // E_ == 1<<15

typedef float v2f __attribute__((ext_vector_type(2)));
typedef float v8f __attribute__((ext_vector_type(8)));

// Order-preserving float -> uint key (for atomicMax-based segment max)
__device__ __forceinline__ unsigned fkey(float f) {
  unsigned u = __float_as_uint(f);
  return (u & 0x80000000u) ? ~u : (u | 0x80000000u);
}
__device__ __forceinline__ float fdec(unsigned k) {
  unsigned u = (k & 0x80000000u) ? (k & 0x7FFFFFFFu) : ~k;
  return __uint_as_float(u);
}

// ---------------------------------------------------------------------------
// Per-layer segment-state init: mkey=0 (below key(-inf)), s=0, agg=0
// grid*block == B*N exactly
// ---------------------------------------------------------------------------
__global__ void k_init(unsigned* __restrict__ mkey, float* __restrict__ s,
                       float* __restrict__ agg) {
  int t = blockIdx.x * blockDim.x + threadIdx.x;
  mkey[2 * t + 0] = 0u;
  mkey[2 * t + 1] = 0u;
  s[2 * t + 0] = 0.0f;
  s[2 * t + 1] = 0.0f;
  float4 z = make_float4(0.f, 0.f, 0.f, 0.f);
  *(float4*)(agg + (size_t)t * 4) = z;
}

// ---------------------------------------------------------------------------
// q = h @ Wq[l]  via V_WMMA_F32_16X16X4_F32 (16 nodes per wave, K=C=4 exact).
// A layout (32-bit, 16x4): lanes 0-15 rows M=0..15 with K={0,1}; lanes 16-31
// same rows with K={2,3}.  B (4x16): VGPR0 = K0/K2 halves, VGPR1 = K1/K3.
// D: VGPR j holds rows j (lanes 0-15) and j+8 (lanes 16-31), col = lane&15.
// Output redistributed through a per-wave LDS tile for coalesced b64 stores.
// ---------------------------------------------------------------------------
__global__ void k_qproj(const float* __restrict__ h, const float* __restrict__ W,
                        float* __restrict__ q) {
  __shared__ float tile[8 * 64];              // 8 waves/block, 16x4 tile each
  float* tl = tile + ((threadIdx.x >> 5) << 6);

  int wave  = blockIdx.x * (blockDim.x >> 5) + (threadIdx.x >> 5);
  int lane  = threadIdx.x & 31;
  int m     = lane & 15;
  int khalf = lane >> 4;
  int base  = wave << 4;

  const float* hr = h + (size_t)(base + m) * 4 + 2 * khalf;
  v2f a; a.x = hr[0]; a.y = hr[1];

  // Unconditional loads (clamped column) + mask-multiply: no EXEC branching.
  int   wc  = m & 3;
  float msk = (m < C_) ? 1.0f : 0.0f;
  v2f bm;
  bm.x = W[(2 * khalf + 0) * C_ + wc] * msk;
  bm.y = W[(2 * khalf + 1) * C_ + wc] * msk;

  v8f c = {0.f, 0.f, 0.f, 0.f, 0.f, 0.f, 0.f, 0.f};
  v8f d = __builtin_amdgcn_wmma_f32_16x16x4_f32(false, a, false, bm,
                                                (short)0, c, false, false);
  if (m < C_) {
#pragma unroll
    for (int j = 0; j < 8; ++j)
      tl[(khalf * 8 + j) * 4 + m] = d[j];
  }
  // Wave-local LDS shuffle: DS ops are in-order within a wave, no barrier.
  v2f r;
  r.x = tl[lane * 2 + 0];
  r.y = tl[lane * 2 + 1];
  *(v2f*)(q + (size_t)base * 4 + lane * 2) = r;
}

// ---------------------------------------------------------------------------
// Edge pass 1: logits + segment max (atomicMax on monotone uint key).
// One edge per lane; msg = [h[src], cos(dt*w+b)]; k = msg @ Wk (8x4).
// ---------------------------------------------------------------------------
__global__ void k_edge_pass1(const float* __restrict__ h, const float* __restrict__ q,
                             const int* __restrict__ edge_index,
                             const float* __restrict__ edge_time,
                             const float* __restrict__ timestamp,
                             const float* __restrict__ time_w,
                             const float* __restrict__ time_b,
                             const float* __restrict__ Wk,
                             unsigned* __restrict__ mkey) {
  int tid = blockIdx.x * blockDim.x + threadIdx.x;
  int b = tid >> LOG2E_;
  int e = tid & (E_ - 1);
  const int* ei = edge_index + (size_t)b * 2 * E_;
  int src = ei[e];
  int dst = ei[E_ + e];

  const float4 hs = *(const float4*)(h + ((size_t)b * N_ + src) * 4);
  float dt = timestamp[b] - edge_time[(size_t)b * E_ + e];

  float msg[8];
  msg[0] = hs.x; msg[1] = hs.y; msg[2] = hs.z; msg[3] = hs.w;
#pragma unroll
  for (int c2 = 0; c2 < TC_; ++c2)
    msg[4 + c2] = __cosf(fmaf(dt, time_w[c2], time_b[c2]));

  float k0 = 0.f, k1 = 0.f, k2 = 0.f, k3 = 0.f;
#pragma unroll
  for (int j = 0; j < 8; ++j) {
    float mj = msg[j];
    k0 = fmaf(mj, Wk[j * 4 + 0], k0);
    k1 = fmaf(mj, Wk[j * 4 + 1], k1);
    k2 = fmaf(mj, Wk[j * 4 + 2], k2);
    k3 = fmaf(mj, Wk[j * 4 + 3], k3);
  }

  const float4 qe = *(const float4*)(q + ((size_t)b * N_ + dst) * 4);
  const float sc = 0.70710678118654752f;  // 1/sqrt(DH)
  float l0 = (qe.x * k0 + qe.y * k1) * sc;
  float l1 = (qe.z * k2 + qe.w * k3) * sc;

  size_t node = (size_t)b * N_ + dst;
  atomicMax(&mkey[node * 2 + 0], fkey(l0));
  atomicMax(&mkey[node * 2 + 1], fkey(l1));
}

// ---------------------------------------------------------------------------
// Edge pass 2: recompute k,v + logits (cheaper than a ~100MB spill; all
// gathers are L2-resident), p = exp(l - m), scatter-add s and p*v.
// ---------------------------------------------------------------------------
__global__ void k_edge_pass2(const float* __restrict__ h, const float* __restrict__ q,
                             const int* __restrict__ edge_index,
                             const float* __restrict__ edge_time,
                             const float* __restrict__ timestamp,
                             const float* __restrict__ time_w,
                             const float* __restrict__ time_b,
                             const float* __restrict__ Wk,
                             const float* __restrict__ Wv,
                             const unsigned* __restrict__ mkey,
                             float* __restrict__ s, float* __restrict__ agg) {
  int tid = blockIdx.x * blockDim.x + threadIdx.x;
  int b = tid >> LOG2E_;
  int e = tid & (E_ - 1);
  const int* ei = edge_index + (size_t)b * 2 * E_;
  int src = ei[e];
  int dst = ei[E_ + e];

  const float4 hs = *(const float4*)(h + ((size_t)b * N_ + src) * 4);
  float dt = timestamp[b] - edge_time[(size_t)b * E_ + e];

  float msg[8];
  msg[0] = hs.x; msg[1] = hs.y; msg[2] = hs.z; msg[3] = hs.w;
#pragma unroll
  for (int c2 = 0; c2 < TC_; ++c2)
    msg[4 + c2] = __cosf(fmaf(dt, time_w[c2], time_b[c2]));

  float k0 = 0.f, k1 = 0.f, k2 = 0.f, k3 = 0.f;
  float v0 = 0.f, v1 = 0.f, v2 = 0.f, v3 = 0.f;
#pragma unroll
  for (int j = 0; j < 8; ++j) {
    float mj = msg[j];
    k0 = fmaf(mj, Wk[j * 4 + 0], k0);
    k1 = fmaf(mj, Wk[j * 4 + 1], k1);
    k2 = fmaf(mj, Wk[j * 4 + 2], k2);
    k3 = fmaf(mj, Wk[j * 4 + 3], k3);
    v0 = fmaf(mj, Wv[j * 4 + 0], v0);
    v1 = fmaf(mj, Wv[j * 4 + 1], v1);
    v2 = fmaf(mj, Wv[j * 4 + 2], v2);
    v3 = fmaf(mj, Wv[j * 4 + 3], v3);
  }

  const float4 qe = *(const float4*)(q + ((size_t)b * N_ + dst) * 4);
  const float sc = 0.70710678118654752f;
  float l0 = (qe.x * k0 + qe.y * k1) * sc;
  float l1 = (qe.z * k2 + qe.w * k3) * sc;

  size_t node = (size_t)b * N_ + dst;
  float m0 = fdec(mkey[node * 2 + 0]);
  float m1 = fdec(mkey[node * 2 + 1]);
  float p0 = __expf(l0 - m0);
  float p1 = __expf(l1 - m1);

  atomicAdd(&s[node * 2 + 0], p0);
  atomicAdd(&s[node * 2 + 1], p1);
  atomicAdd(&agg[node * 4 + 0], p0 * v0);
  atomicAdd(&agg[node * 4 + 1], p0 * v1);
  atomicAdd(&agg[node * 4 + 2], p1 * v2);
  atomicAdd(&agg[node * 4 + 3], p1 * v3);
}

// ---------------------------------------------------------------------------
// Output projection: attn = agg/s (s==0 -> 1), h' = relu(h + attn@Wo + bo),
// same 16x16x4 f32 WMMA tiling; LDS redistribution for coalesced I/O.
// ---------------------------------------------------------------------------
__global__ void k_outproj(const float* __restrict__ h_in, const float* __restrict__ agg,
                          const float* __restrict__ s, const float* __restrict__ Wo,
                          const float* __restrict__ bo, float* __restrict__ h_out) {
  __shared__ float tile[8 * 64];
  float* tl = tile + ((threadIdx.x >> 5) << 6);

  int wave  = blockIdx.x * (blockDim.x >> 5) + (threadIdx.x >> 5);
  int lane  = threadIdx.x & 31;
  int m     = lane & 15;
  int khalf = lane >> 4;
  int base  = wave << 4;
  int row   = base + m;

  // attn components c = 2*khalf, 2*khalf+1 both belong to head khalf
  float sv = s[(size_t)row * 2 + khalf];
  sv = (sv == 0.0f) ? 1.0f : sv;
  float inv = 1.0f / sv;
  const float* ag = agg + (size_t)row * 4 + 2 * khalf;
  v2f a; a.x = ag[0] * inv; a.y = ag[1] * inv;

  int   wc  = m & 3;
  float msk = (m < C_) ? 1.0f : 0.0f;
  v2f bm;
  bm.x = Wo[(2 * khalf + 0) * C_ + wc] * msk;
  bm.y = Wo[(2 * khalf + 1) * C_ + wc] * msk;

  v8f c = {0.f, 0.f, 0.f, 0.f, 0.f, 0.f, 0.f, 0.f};
  v8f d = __builtin_amdgcn_wmma_f32_16x16x4_f32(false, a, false, bm,
                                                (short)0, c, false, false);
  if (m < C_) {
#pragma unroll
    for (int j = 0; j < 8; ++j)
      tl[(khalf * 8 + j) * 4 + m] = d[j];
  }
  // Wave-local LDS shuffle; then fully coalesced residual + relu + store.
  int c0 = (lane * 2 + 0) & 3;
  int c1 = (lane * 2 + 1) & 3;
  const float* hrow = h_in + (size_t)base * 4 + lane * 2;
  float o0 = hrow[0] + tl[lane * 2 + 0] + bo[c0];
  float o1 = hrow[1] + tl[lane * 2 + 1] + bo[c1];
  o0 = o0 > 0.0f ? o0 : 0.0f;
  o1 = o1 > 0.0f ? o1 : 0.0f;
  v2f st; st.x = o0; st.y = o1;
  *(v2f*)(h_out + (size_t)base * 4 + lane * 2) = st;
}

// ---------------------------------------------------------------------------
// Readout: feats(12) @ W_lin(12x2) + b_lin ; one thread per batch.
// ---------------------------------------------------------------------------
__global__ void k_final(const float* __restrict__ h2, const int* __restrict__ src_index,
                        const int* __restrict__ dst_index,
                        const float* __restrict__ timestamp,
                        const float* __restrict__ time_w, const float* __restrict__ time_b,
                        const float* __restrict__ W_lin, const float* __restrict__ b_lin,
                        float* __restrict__ out) {
  int b = blockIdx.x * blockDim.x + threadIdx.x;
  if (b >= B_) return;
  const float* sx = h2 + ((size_t)b * N_ + src_index[b]) * 4;
  const float* dx = h2 + ((size_t)b * N_ + dst_index[b]) * 4;
  float ts = timestamp[b];
  float f[12];
#pragma unroll
  for (int c2 = 0; c2 < 4; ++c2) {
    f[c2]     = sx[c2];
    f[4 + c2] = dx[c2];
    f[8 + c2] = __cosf(fmaf(ts, time_w[c2], time_b[c2]));
  }
#pragma unroll
  for (int o = 0; o < 2; ++o) {
    float acc = b_lin[o];
#pragma unroll
    for (int j = 0; j < 12; ++j) acc = fmaf(f[j], W_lin[j * 2 + o], acc);
    out[b * 2 + o] = acc;
  }
}

// ---------------------------------------------------------------------------
extern "C" void kernel_launch(void* const* d_in, const int* in_sizes, int n_in,
                              void* d_out, int out_size, void* d_ws, size_t ws_size,
                              hipStream_t stream) {
  const float* x          = (const float*)d_in[0];
  const int*   edge_index = (const int*)d_in[1];
  const float* edge_time  = (const float*)d_in[2];
  const float* timestamp  = (const float*)d_in[3];
  const int*   src_index  = (const int*)d_in[4];
  const int*   dst_index  = (const int*)d_in[5];
  const float* time_w     = (const float*)d_in[6];
  const float* time_b     = (const float*)d_in[7];
  const float* Wq         = (const float*)d_in[8];
  const float* Wk         = (const float*)d_in[9];
  const float* Wv         = (const float*)d_in[10];
  const float* Wo         = (const float*)d_in[11];
  const float* bo         = (const float*)d_in[12];
  const float* W_lin      = (const float*)d_in[13];
  const float* b_lin      = (const float*)d_in[14];
  float* out = (float*)d_out;

  // Workspace layout (~21 MB total)
  const size_t HN = (size_t)B_ * N_ * C_;  // 1,048,576 floats
  const size_t SN = (size_t)B_ * N_ * H_;  //   524,288 floats
  float*    ws   = (float*)d_ws;
  float*    h1   = ws;
  float*    h2   = h1 + HN;
  float*    q    = h2 + HN;
  unsigned* mkey = (unsigned*)(q + HN);
  float*    s    = (float*)(mkey + SN);
  float*    agg  = s + SN;

  const int init_blocks = (B_ * N_) / 256;        // 1024
  const int edge_blocks = (B_ * E_) / 256;        // 16384
  const int node_blocks = ((B_ * N_) / 16) / 8;   // 2048 (8 waves/block, 16 rows/wave)

  for (int l = 0; l < L_; ++l) {
    const float* h_in  = (l == 0) ? x : h1;
    float*       h_out = (l == 0) ? h1 : h2;
    const float* Wql = Wq + l * C_ * C_;
    const float* Wkl = Wk + l * (C_ + TC_) * C_;
    const float* Wvl = Wv + l * (C_ + TC_) * C_;
    const float* Wol = Wo + l * C_ * C_;
    const float* bol = bo + l * C_;

    k_init<<<init_blocks, 256, 0, stream>>>(mkey, s, agg);
    k_qproj<<<node_blocks, 256, 0, stream>>>(h_in, Wql, q);
    k_edge_pass1<<<edge_blocks, 256, 0, stream>>>(h_in, q, edge_index, edge_time,
                                                  timestamp, time_w, time_b, Wkl, mkey);
    k_edge_pass2<<<edge_blocks, 256, 0, stream>>>(h_in, q, edge_index, edge_time,
                                                  timestamp, time_w, time_b, Wkl, Wvl,
                                                  mkey, s, agg);
    k_outproj<<<node_blocks, 256, 0, stream>>>(h_in, agg, s, Wol, bol, h_out);
  }
  k_final<<<1, 128, 0, stream>>>(h2, src_index, dst_index, timestamp,
                                 time_w, time_b, W_lin, b_lin, out);
}